// ScalarRoPEEmbedding_83769042141635
// MI455X (gfx1250) — compile-verified
//
#include <hip/hip_runtime.h>

// RoPE sin/cos table gather + pair swap.
// out[row, 2k]   = cache[pos[row], k, 1]  (cos)
// out[row, 2k+1] = cache[pos[row], k, 0]  (sin)
//
// Memory-bound: ~128 MiB NT stores + ~20 MB L2-resident table reads
// (floor ≈ 6.7 us @ 23.3 TB/s). Zero FLOPs -> WMMA not applicable;
// the gfx1250-relevant paths are data movement:
//   s_load_b32 (wave-uniform position)  -> SMEM/KMcnt
//   global_load_b128                    -> gathered cache row (L2 hit)
//   v shuffle [1,0,3,2]                 -> pair swap, free VALU
//   global_store_b128 TH=NT             -> keep table resident in 192MB L2
//   global_prefetch_b8                  -> next row, software pipeline

typedef __attribute__((ext_vector_type(4))) float v4f;

#define ROPE_ROWS      (8 * 8192)   // BATCH * SEQ
#define ROPE_DIM       512          // floats per row
#define ROW_F4         128          // float4 chunks per row
#define THREADS        256          // 8 wave32 / block; 2 rows per block-iter
#define ROWS_PER_BLK   2
#define ITERS          8
#define BLOCKS         (ROPE_ROWS / (ROWS_PER_BLK * ITERS))  // 4096

__global__ __launch_bounds__(THREADS)
void ScalarRoPEEmbedding_83769042141635_kernel(const int*   __restrict__ positions,
                                               const float* __restrict__ cache,
                                               float*       __restrict__ out)
{
    const int chunk = threadIdx.x & (ROW_F4 - 1);   // float4 index within row
    const int sub   = threadIdx.x >> 7;             // which of the 2 rows (wave-uniform)
    const int row_stride = gridDim.x * ROWS_PER_BLK;

    int row = blockIdx.x * ROWS_PER_BLK + sub;
    if (row >= ROPE_ROWS) return;

    // row is uniform across each wave32 -> force scalar (s_load) path.
    int urow = __builtin_amdgcn_readfirstlane(row);
    int pos  = positions[urow];

    #pragma unroll 1
    for (int it = 0; it < ITERS; ++it) {
        const int next_row = row + row_stride;
        int next_pos = 0;
        if ((it + 1 < ITERS) && (next_row < ROPE_ROWS)) {
            int unext = __builtin_amdgcn_readfirstlane(next_row);
            next_pos  = positions[unext];            // s_load_b32 (KMcnt)
            // gfx1250 prefetch: pull next gathered row toward the WGP
            __builtin_prefetch(cache + (size_t)next_pos * ROPE_DIM + chunk * 4, 0, 1);
        }

        // {sin0, cos0, sin1, cos1} -> {cos0, sin0, cos1, sin1}
        const v4f* src = (const v4f*)(cache + (size_t)pos * ROPE_DIM) + chunk;
        v4f v = *src;                                         // global_load_b128
        v4f w = __builtin_shufflevector(v, v, 1, 0, 3, 2);

        // Non-temporal store: 128 MiB output stream must not evict the
        // 20 MB sin/cos table from the 192 MB L2.
        v4f* dst = (v4f*)(out + (size_t)row * ROPE_DIM) + chunk;
        __builtin_nontemporal_store(w, dst);                  // global_store_b128 TH=NT

        row = next_row;
        pos = next_pos;
    }
}

extern "C" void kernel_launch(void* const* d_in, const int* in_sizes, int n_in,
                              void* d_out, int out_size, void* d_ws, size_t ws_size,
                              hipStream_t stream) {
    (void)in_sizes; (void)n_in; (void)d_ws; (void)ws_size; (void)out_size;
    const int*   positions = (const int*)d_in[0];    // [8, 8192]
    const float* cache     = (const float*)d_in[1];  // [10000, 256, 2]
    float*       out       = (float*)d_out;          // [8, 8192, 512]

    ScalarRoPEEmbedding_83769042141635_kernel<<<BLOCKS, THREADS, 0, stream>>>(
        positions, cache, out);
}